// GCN_43533788512980
// MI455X (gfx1250) — compile-verified
//
#include <hip/hip_runtime.h>
#include <hip/hip_bf16.h>

typedef __attribute__((ext_vector_type(2))) float v2f;
typedef __attribute__((ext_vector_type(8))) float v8f;

// ---------------------------------------------------------------------------
// Degree / normalization kernels
// ---------------------------------------------------------------------------
__global__ void deg_init_kernel(float* __restrict__ deg, int n) {
    int i = blockIdx.x * blockDim.x + threadIdx.x;
    int stride = gridDim.x * blockDim.x;
    for (; i < n; i += stride) deg[i] = 1.0f;  // self-loop contributes 1
}

__global__ void deg_edges_kernel(float* __restrict__ deg, const int* __restrict__ dst, int e) {
    int i = blockIdx.x * blockDim.x + threadIdx.x;
    int stride = gridDim.x * blockDim.x;
    for (; i < e; i += stride) atomicAdd(&deg[dst[i]], 1.0f);
}

__global__ void dinv_kernel(float* __restrict__ deg, int n) {
    int i = blockIdx.x * blockDim.x + threadIdx.x;
    int stride = gridDim.x * blockDim.x;
    for (; i < n; i += stride) deg[i] = __frsqrt_rn(deg[i]);  // deg >= 1 always
}

// ---------------------------------------------------------------------------
// GEMM: H[M x FOUT] = X[M x K] @ W[K x FOUT], exact fp32 via V_WMMA_F32_16X16X4_F32.
// One wave owns a 16-row strip of X and all FOUT/16 column tiles, so X is
// streamed from HBM exactly once; W stays hot in WGP$/L2.
//
// A fragment (16x4 f32): lanes 0-15 -> M=lane, K={0,1} in v.x/v.y;
//                        lanes 16-31 -> same M, K={2,3}.
// B fragment (4x16 f32): lanes index N, same K split.
// C/D (16x16 f32): VGPR j holds M = j + 8*(lane>=16), N = lane&15.
// ---------------------------------------------------------------------------
template <int BN>  // BN = FOUT/16
__launch_bounds__(256)
__global__ void gemm_wmma_f32_kernel(const float* __restrict__ X,
                                     const float* __restrict__ W,
                                     float* __restrict__ H,
                                     int M, int K) {
    constexpr int FOUT = BN * 16;
    const int lane  = threadIdx.x & 31;
    const int strip = blockIdx.x * (blockDim.x >> 5) + (threadIdx.x >> 5);
    if (strip * 16 >= M) return;                 // wave-uniform: EXEC stays full

    const int half = lane >> 4;                  // 0 or 1
    const int l15  = lane & 15;
    const int klo  = half * 2;                   // K sub-offset {0,2}
    const float* __restrict__ xrow = X + (size_t)(strip * 16 + l15) * K;

    v8f acc[BN];
#pragma unroll
    for (int t = 0; t < BN; ++t) acc[t] = (v8f){0.f,0.f,0.f,0.f,0.f,0.f,0.f,0.f};

    for (int k = 0; k < K; k += 4) {
        // 8-byte aligned float2 load of the A fragment
        const v2f a = *reinterpret_cast<const v2f*>(xrow + k + klo);
        const float* wr0 = W + (size_t)(k + klo) * FOUT + l15;
#pragma unroll
        for (int t = 0; t < BN; ++t) {
            v2f b;
            b.x = wr0[t * 16];
            b.y = wr0[t * 16 + FOUT];
            acc[t] = __builtin_amdgcn_wmma_f32_16x16x4_f32(
                false, a, false, b, (short)0, acc[t], false, false);
        }
    }

    const int mBase = strip * 16 + half * 8;
#pragma unroll
    for (int t = 0; t < BN; ++t) {
#pragma unroll
        for (int j = 0; j < 8; ++j) {
            H[(size_t)(mBase + j) * FOUT + t * 16 + l15] = acc[t][j];
        }
    }
}

// ---------------------------------------------------------------------------
// OUT[i,:] = dinv[i]^2 * H[i,:]   (self-loop term; also initializes OUT)
// ---------------------------------------------------------------------------
__global__ void self_init_kernel(const float* __restrict__ H,
                                 const float* __restrict__ dinv,
                                 float* __restrict__ OUT, int n, int f) {
    size_t total4 = (size_t)n * f / 4;
    size_t stride = (size_t)gridDim.x * blockDim.x;
    for (size_t i = (size_t)blockIdx.x * blockDim.x + threadIdx.x; i < total4; i += stride) {
        int node = (int)((i * 4) / (unsigned)f);
        float w = dinv[node];
        w *= w;
        float4 v = reinterpret_cast<const float4*>(H)[i];
        float4 o;
        o.x = w * v.x; o.y = w * v.y; o.z = w * v.z; o.w = w * v.w;
        reinterpret_cast<float4*>(OUT)[i] = o;
    }
}

// ---------------------------------------------------------------------------
// OUT[dst[e],:] += dinv[src]*dinv[dst] * H[src[e],:]
// One wave per edge: coalesced float4 gather of the source row,
// global_atomic_add_f32 scatter to the destination row.
// ---------------------------------------------------------------------------
__launch_bounds__(256)
__global__ void edge_aggregate_kernel(const float* __restrict__ H,
                                      const int* __restrict__ src,
                                      const int* __restrict__ dst,
                                      const float* __restrict__ dinv,
                                      float* __restrict__ OUT,
                                      int nEdges, int f) {
    const int lane   = threadIdx.x & 31;
    const int nWaves = (gridDim.x * blockDim.x) >> 5;
    for (int e = (blockIdx.x * blockDim.x + threadIdx.x) >> 5; e < nEdges; e += nWaves) {
        const int s = src[e];
        const int d = dst[e];
        const float nrm = dinv[s] * dinv[d];
        const float4* hs = reinterpret_cast<const float4*>(H + (size_t)s * f);
        float* od = OUT + (size_t)d * f;
        for (int f4 = lane; f4 * 4 < f; f4 += 32) {
            float4 v = hs[f4];
            int fb = f4 * 4;
            atomicAdd(od + fb + 0, nrm * v.x);
            atomicAdd(od + fb + 1, nrm * v.y);
            atomicAdd(od + fb + 2, nrm * v.z);
            atomicAdd(od + fb + 3, nrm * v.w);
        }
    }
}

// ---------------------------------------------------------------------------
// OUT = relu(OUT + bias)   (f is a power of two)
// ---------------------------------------------------------------------------
__global__ void bias_relu_kernel(float* __restrict__ OUT, const float* __restrict__ b,
                                 int n, int f) {
    size_t total  = (size_t)n * f;
    size_t stride = (size_t)gridDim.x * blockDim.x;
    for (size_t i = (size_t)blockIdx.x * blockDim.x + threadIdx.x; i < total; i += stride) {
        float v = OUT[i] + b[i & (size_t)(f - 1)];
        OUT[i] = v > 0.0f ? v : 0.0f;
    }
}

// ---------------------------------------------------------------------------
// Segment-max pool (F == 32, one lane per feature). ReLU outputs are >= 0,
// so float max == unsigned bit-pattern max. Batch is sorted -> run-length
// local max, atomicMax only at segment boundaries.
// ---------------------------------------------------------------------------
__global__ void pool_init_kernel(unsigned* __restrict__ pooled, int n) {
    int i = blockIdx.x * blockDim.x + threadIdx.x;
    if (i < n) pooled[i] = 0u;  // bit pattern of +0.0f
}

__launch_bounds__(256)
__global__ void pool_max_kernel(const float* __restrict__ H,
                                const int* __restrict__ batch,
                                unsigned* __restrict__ pooled,
                                int n, int chunk) {
    const int lane = threadIdx.x & 31;
    const int wave = (blockIdx.x * blockDim.x + threadIdx.x) >> 5;
    const int start = wave * chunk;
    if (start >= n) return;
    const int end = min(start + chunk, n);
    int curg = batch[start];
    float m = 0.0f;
    for (int i = start; i < end; ++i) {
        int g = batch[i];
        if (g != curg) {
            atomicMax(&pooled[curg * 32 + lane], __float_as_uint(m));
            m = 0.0f;
            curg = g;
        }
        m = fmaxf(m, H[(size_t)i * 32 + lane]);
    }
    atomicMax(&pooled[curg * 32 + lane], __float_as_uint(m));
}

// ---------------------------------------------------------------------------
// logits = pooled @ Wfc + bfc ; log_softmax.  64 graphs, one thread each.
// ---------------------------------------------------------------------------
__global__ void head_kernel(const float* __restrict__ pooled,
                            const float* __restrict__ Wfc,
                            const float* __restrict__ bfc,
                            float* __restrict__ out, int nGraphs) {
    int g = blockIdx.x * blockDim.x + threadIdx.x;
    if (g >= nGraphs) return;
    float logit[10];
#pragma unroll
    for (int c = 0; c < 10; ++c) logit[c] = bfc[c];
    for (int k = 0; k < 32; ++k) {
        float p = pooled[g * 32 + k];
#pragma unroll
        for (int c = 0; c < 10; ++c) logit[c] += p * Wfc[k * 10 + c];
    }
    float m = logit[0];
#pragma unroll
    for (int c = 1; c < 10; ++c) m = fmaxf(m, logit[c]);
    float s = 0.0f;
#pragma unroll
    for (int c = 0; c < 10; ++c) s += __expf(logit[c] - m);
    float lse = __logf(s);
#pragma unroll
    for (int c = 0; c < 10; ++c) out[g * 10 + c] = logit[c] - m - lse;
}

// ---------------------------------------------------------------------------
// Launch
// ---------------------------------------------------------------------------
extern "C" void kernel_launch(void* const* d_in, const int* in_sizes, int n_in,
                              void* d_out, int out_size, void* d_ws, size_t ws_size,
                              hipStream_t stream) {
    const float* x     = (const float*)d_in[0];
    const int*   ei    = (const int*)d_in[1];
    const int*   batch = (const int*)d_in[2];
    const float* W1    = (const float*)d_in[3];
    const float* b1    = (const float*)d_in[4];
    const float* W2    = (const float*)d_in[5];
    const float* b2    = (const float*)d_in[6];
    const float* W3    = (const float*)d_in[7];
    const float* b3    = (const float*)d_in[8];
    const float* Wfc   = (const float*)d_in[9];
    const float* bfc   = (const float*)d_in[10];

    const int N = in_sizes[2];          // 100000 (multiple of 16)
    const int E = in_sizes[1] / 2;      // 1600000
    const int G = 64;
    const int* src = ei;                // edge_index[0]
    const int* dst = ei + E;            // edge_index[1]

    // Workspace layout (~103 MB): dinv | buf1(N*128) | buf2(N*128) | pooled
    float* dinv = (float*)d_ws;
    float* buf1 = dinv + N;
    float* buf2 = buf1 + (size_t)N * 128;
    unsigned* pooled = (unsigned*)(buf2 + (size_t)N * 128);

    const int strips      = N / 16;                 // 6250
    const int gemmBlocks  = (strips + 7) / 8;       // 8 waves/block

    // --- normalization ---
    deg_init_kernel<<<(N + 255) / 256, 256, 0, stream>>>(dinv, N);
    deg_edges_kernel<<<2048, 256, 0, stream>>>(dinv, dst, E);
    dinv_kernel<<<(N + 255) / 256, 256, 0, stream>>>(dinv, N);

    // --- layer 1: 512 -> 128 ---
    gemm_wmma_f32_kernel<8><<<gemmBlocks, 256, 0, stream>>>(x, W1, buf1, N, 512);
    self_init_kernel<<<2048, 256, 0, stream>>>(buf1, dinv, buf2, N, 128);
    edge_aggregate_kernel<<<4096, 256, 0, stream>>>(buf1, src, dst, dinv, buf2, E, 128);
    bias_relu_kernel<<<2048, 256, 0, stream>>>(buf2, b1, N, 128);

    // --- layer 2: 128 -> 64 ---
    gemm_wmma_f32_kernel<4><<<gemmBlocks, 256, 0, stream>>>(buf2, W2, buf1, N, 128);
    self_init_kernel<<<2048, 256, 0, stream>>>(buf1, dinv, buf2, N, 64);
    edge_aggregate_kernel<<<4096, 256, 0, stream>>>(buf1, src, dst, dinv, buf2, E, 64);
    bias_relu_kernel<<<2048, 256, 0, stream>>>(buf2, b2, N, 64);

    // --- layer 3: 64 -> 32 ---
    gemm_wmma_f32_kernel<2><<<gemmBlocks, 256, 0, stream>>>(buf2, W3, buf1, N, 64);
    self_init_kernel<<<1024, 256, 0, stream>>>(buf1, dinv, buf2, N, 32);
    edge_aggregate_kernel<<<4096, 256, 0, stream>>>(buf1, src, dst, dinv, buf2, E, 32);
    bias_relu_kernel<<<1024, 256, 0, stream>>>(buf2, b3, N, 32);

    // --- global max pool + head ---
    pool_init_kernel<<<(G * 32 + 255) / 256, 256, 0, stream>>>(pooled, G * 32);
    const int chunk = 128;
    const int poolWaves  = (N + chunk - 1) / chunk;
    const int poolBlocks = (poolWaves + 7) / 8;
    pool_max_kernel<<<poolBlocks, 256, 0, stream>>>(buf2, batch, pooled, N, chunk);
    head_kernel<<<1, 64, 0, stream>>>((const float*)pooled, Wfc, bfc, (float*)d_out, G);
}